// EPCLSegNet_58394375357153
// MI455X (gfx1250) — compile-verified
//
#include <hip/hip_runtime.h>
#include <math.h>

// ---------------------------------------------------------------------------
// PointMixerBlock for MI455X (gfx1250, wave32, WMMA).
// Pipeline:
//   zero_stats
//   gemm64_bn: t1 = x@W1                      (+stats1)
//   gemm64_bn: v  = relu(BN1(t1))@Wv
//   mixer    : mixed = intra-set mixing        (+stats2)   [wave-per-point WMMA]
//   gemm64_bn: t3 = relu(BN2(mixed))@W3        (+stats3)
//   final    : out = relu(BN3(t3) + x)
// ---------------------------------------------------------------------------

typedef float v2f __attribute__((ext_vector_type(2)));
typedef float v8f __attribute__((ext_vector_type(8)));

#define BN_EPS 1e-5f

// D = A(16x4 f32) * B(4x16 f32) + C, one matrix per wave32.
// Fragment layouts per CDNA5 ISA 7.12.2:
//   A/B: element K = 2*(lane>>4) + reg, M/N = lane&15   (2 VGPRs)
//   C/D: element M = reg + 8*(lane>>4), N = lane&15     (8 VGPRs)
__device__ __forceinline__ v8f wmma_f32(v2f a, v2f b, v8f c) {
    return __builtin_amdgcn_wmma_f32_16x16x4_f32(
        /*neg_a=*/false, a, /*neg_b=*/false, b,
        /*c_mod=*/(short)0, c, /*reuse_a=*/false, /*reuse_b=*/false);
}

__global__ void zero_stats_kernel(float* stats, int n) {
    int i = blockIdx.x * blockDim.x + threadIdx.x;
    if (i < n) stats[i] = 0.f;
}

// ---------------------------------------------------------------------------
// Out[N,64] = f(A)[N,64] @ W[64,64], f = optional BN(statsIn,gamma,beta)+ReLU.
// Optionally accumulates per-channel sum / sumsq of Out into statsOut[128].
// 128 threads = 4 waves; 64 rows per block; wave w owns rows 16w..16w+15.
// ---------------------------------------------------------------------------
__global__ __launch_bounds__(128) void gemm64_bn_kernel(
    const float* __restrict__ A, const float* __restrict__ W,
    float* __restrict__ Out, int N,
    const float* __restrict__ statsIn, const float* __restrict__ gamma,
    const float* __restrict__ beta, float* __restrict__ statsOut)
{
    __shared__ float As[64][64];
    __shared__ float Wsh[64][64];
    __shared__ float sc[64], bi[64];
    __shared__ float sacc[128];

    const int tid  = threadIdx.x;
    const int lane = tid & 31;
    const int wv   = tid >> 5;
    const int half = lane >> 4;
    const int l16  = lane & 15;
    const int row0 = blockIdx.x * 64;

    if (statsIn && tid < 64) {
        float mean = statsIn[tid] / (float)N;
        float var  = statsIn[64 + tid] / (float)N - mean * mean;
        float s    = gamma[tid] * rsqrtf(var + BN_EPS);
        sc[tid] = s;
        bi[tid] = beta[tid] - mean * s;
    }
    if (statsOut && tid < 128) sacc[tid] = 0.f;
    for (int i = tid; i < 64 * 64; i += 128) Wsh[i >> 6][i & 63] = W[i];
    __syncthreads();   // sc/bi ready for A-load

    for (int i = tid; i < 64 * 64; i += 128) {
        int r = i >> 6, c = i & 63;
        int gr = row0 + r;
        float v = (gr < N) ? A[(size_t)gr * 64 + c] : 0.f;
        if (statsIn) v = fmaxf(sc[c] * v + bi[c], 0.f);
        As[r][c] = v;
    }
    __syncthreads();

    const int mrow = wv * 16 + l16;
#pragma unroll
    for (int nb = 0; nb < 4; ++nb) {
        v8f acc = {};
#pragma unroll
        for (int t = 0; t < 16; ++t) {
            int k0 = 4 * t + 2 * half;
            v2f af = { As[mrow][k0], As[mrow][k0 + 1] };
            v2f bf = { Wsh[k0][nb * 16 + l16], Wsh[k0 + 1][nb * 16 + l16] };
            acc = wmma_f32(af, bf, acc);
        }
        int c = nb * 16 + l16;
        float lsum = 0.f, lssq = 0.f;
#pragma unroll
        for (int r = 0; r < 8; ++r) {
            int grow = row0 + wv * 16 + r + 8 * half;
            if (grow < N) {
                float v = acc[r];
                Out[(size_t)grow * 64 + c] = v;
                lsum += v; lssq += v * v;
            }
        }
        if (statsOut) {
            atomicAdd(&sacc[c], lsum);          // ds_add_f32
            atomicAdd(&sacc[64 + c], lssq);
        }
    }
    if (statsOut) {
        __syncthreads();
        if (tid < 128) atomicAdd(&statsOut[tid], sacc[tid]);
    }
}

// ---------------------------------------------------------------------------
// Intra-set mixer: one wave32 per point; the K=16 neighbors ARE the M=16
// dimension of the WMMA tiles. 4 waves (4 points) per 128-thread block.
// ---------------------------------------------------------------------------
__global__ __launch_bounds__(128) void mixer_kernel(
    const float* __restrict__ p, const int* __restrict__ knn,
    const float* __restrict__ t1, const float* __restrict__ vbuf,
    const float* __restrict__ Wp1, const float* __restrict__ Wp2,
    const float* __restrict__ Ww,
    const float* __restrict__ stats1, const float* __restrict__ g1,
    const float* __restrict__ b1,
    float* __restrict__ mixed, float* __restrict__ statsOut, int N)
{
    __shared__ float Wp2s[64][64];       // 16 KB
    __shared__ float Wp1s[4][64];        //  1 KB (row 3 = 0 pad)
    __shared__ float Wws[64][16];        //  4 KB (cols 8..15 = 0 pad)
    __shared__ float sc1[64], bi1[64];
    __shared__ float sacc[128];
    __shared__ float hjs[4][16][64];     // 16 KB: gathered h (BN1+relu applied)
    __shared__ float es [4][16][64];     // 16 KB: u then e, in row-major
    __shared__ float wsft[4][16][8];     //  2 KB: softmax weights

    const int tid  = threadIdx.x;
    const int lane = tid & 31;
    const int wv   = tid >> 5;
    const int half = lane >> 4;
    const int l16  = lane & 15;
    const int n    = blockIdx.x * 4 + wv;

    if (tid < 64) {
        float mean = stats1[tid] / (float)N;
        float var  = stats1[64 + tid] / (float)N - mean * mean;
        float s    = g1[tid] * rsqrtf(var + BN_EPS);
        sc1[tid] = s; bi1[tid] = b1[tid] - mean * s;
    }
    if (tid < 128) sacc[tid] = 0.f;
    for (int i = tid; i < 64 * 64; i += 128) Wp2s[i >> 6][i & 63] = Wp2[i];
    for (int i = tid; i < 4 * 64; i += 128) {
        int r = i >> 6, c = i & 63;
        Wp1s[r][c] = (r < 3) ? Wp1[r * 64 + c] : 0.f;
    }
    for (int i = tid; i < 64 * 16; i += 128) {
        int r = i >> 4, c = i & 15;
        Wws[r][c] = (c < 8) ? Ww[r * 8 + c] : 0.f;
    }
    __syncthreads();

    if (n < N) {
        // --- neighbor indices (lanes 0..15 own one each) & relative coords ---
        int jreg = (lane < 16) ? knn[n * 16 + lane] : 0;
        float pnx = p[3 * n + 0], pny = p[3 * n + 1], pnz = p[3 * n + 2];
        float dx = 0.f, dy = 0.f, dz = 0.f;
        if (lane < 16) {
            dx = p[3 * jreg + 0] - pnx;
            dy = p[3 * jreg + 1] - pny;
            dz = p[3 * jreg + 2] - pnz;
        }

        // --- stage hj = relu(BN1(t1[j])) into LDS (coalesced 64B segments) ---
#pragma unroll
        for (int m = 0; m < 16; ++m) {
            int jm = __shfl(jreg, m, 32);
            const float* src = t1 + (size_t)jm * 64;
            float a = src[lane];
            float b = src[lane + 32];
            hjs[wv][m][lane]      = fmaxf(sc1[lane]      * a + bi1[lane],      0.f);
            hjs[wv][m][lane + 32] = fmaxf(sc1[lane + 32] * b + bi1[lane + 32], 0.f);
        }

        // --- A fragment of delta (16x4, col 3 zero) built via shuffles ---
        float a0 = __shfl(dx, l16, 32);
        float a1 = __shfl(dy, l16, 32);
        float a2 = __shfl(dz, l16, 32);
        v2f adel;
        adel.x = half ? a2  : a0;
        adel.y = half ? 0.f : a1;

        // --- GEMM1: u = relu(delta @ Wp1)  ->  es (row-major for A-frags) ---
#pragma unroll
        for (int nb = 0; nb < 4; ++nb) {
            v2f bf = { Wp1s[2 * half][nb * 16 + l16],
                       Wp1s[2 * half + 1][nb * 16 + l16] };
            v8f u = {};
            u = wmma_f32(adel, bf, u);
#pragma unroll
            for (int r = 0; r < 8; ++r)
                es[wv][r + 8 * half][nb * 16 + l16] = fmaxf(u[r], 0.f);
        }
        __builtin_amdgcn_wave_barrier();
        asm volatile("s_wait_dscnt 0" ::: "memory");

        // --- GEMM2: e = u @ Wp2 (accumulators stay in VGPRs) ---
        v8f eacc[4];
#pragma unroll
        for (int nb = 0; nb < 4; ++nb) {
            v8f acc = {};
#pragma unroll
            for (int t = 0; t < 16; ++t) {
                int k0 = 4 * t + 2 * half;
                v2f af = { es[wv][l16][k0], es[wv][l16][k0 + 1] };
                v2f bf = { Wp2s[k0][nb * 16 + l16], Wp2s[k0 + 1][nb * 16 + l16] };
                acc = wmma_f32(af, bf, acc);
            }
            eacc[nb] = acc;
        }
        __builtin_amdgcn_wave_barrier();
        // u fully consumed -> overwrite es with e (D-layout -> row-major)
#pragma unroll
        for (int nb = 0; nb < 4; ++nb)
#pragma unroll
            for (int r = 0; r < 8; ++r)
                es[wv][r + 8 * half][nb * 16 + l16] = eacc[nb][r];
        __builtin_amdgcn_wave_barrier();
        asm volatile("s_wait_dscnt 0" ::: "memory");

        // --- GEMM3: energy = (hj + e) @ Ww  (N padded 8->16 with zeros) ---
        v8f en = {};
#pragma unroll
        for (int t = 0; t < 16; ++t) {
            int k0 = 4 * t + 2 * half;
            v2f af = { hjs[wv][l16][k0]     + es[wv][l16][k0],
                       hjs[wv][l16][k0 + 1] + es[wv][l16][k0 + 1] };
            v2f bf = { Wws[k0][l16], Wws[k0 + 1][l16] };
            en = wmma_f32(af, bf, en);
        }

        // --- softmax over the 16 neighbors (8 regs + cross-half shuffle) ---
        float mx = en[0];
#pragma unroll
        for (int r = 1; r < 8; ++r) mx = fmaxf(mx, en[r]);
        mx = fmaxf(mx, __shfl_xor(mx, 16, 32));
        float ex[8];
        float sum = 0.f;
#pragma unroll
        for (int r = 0; r < 8; ++r) { ex[r] = __expf(en[r] - mx); sum += ex[r]; }
        sum += __shfl_xor(sum, 16, 32);
        float inv = 1.f / sum;
        if (l16 < 8) {
#pragma unroll
            for (int r = 0; r < 8; ++r) wsft[wv][r + 8 * half][l16] = ex[r] * inv;
        }
        __builtin_amdgcn_wave_barrier();
        asm volatile("s_wait_dscnt 0" ::: "memory");

        // --- mixed[c] = sum_m w[m][c>>3] * (v[j_m][c] + e[m][c]) ---
        int jm[8];
#pragma unroll
        for (int r = 0; r < 8; ++r) jm[r] = __shfl(jreg, r + 8 * half, 32);
#pragma unroll
        for (int nb = 0; nb < 4; ++nb) {
            int c = nb * 16 + l16;
            int g = 2 * nb + (l16 >> 3);   // channel group, 0..7
            float ps = 0.f;
#pragma unroll
            for (int r = 0; r < 8; ++r) {
                float wgt = wsft[wv][r + 8 * half][g];
                ps += wgt * (vbuf[(size_t)jm[r] * 64 + c] + eacc[nb][r]);
            }
            ps += __shfl_xor(ps, 16, 32);
            if (half == 0) {
                mixed[(size_t)n * 64 + c] = ps;
                atomicAdd(&sacc[c], ps);
                atomicAdd(&sacc[64 + c], ps * ps);
            }
        }
    }
    __syncthreads();
    if (tid < 128) atomicAdd(&statsOut[tid], sacc[tid]);
}

// out = relu(BN3(t3) + x)
__global__ void final_kernel(const float* __restrict__ t3,
                             const float* __restrict__ x,
                             const float* __restrict__ stats3,
                             const float* __restrict__ g3,
                             const float* __restrict__ b3,
                             float* __restrict__ out, int N)
{
    int i = blockIdx.x * blockDim.x + threadIdx.x;
    if (i < N * 64) {
        int c = i & 63;
        float mean = stats3[c] / (float)N;
        float var  = stats3[64 + c] / (float)N - mean * mean;
        float s    = g3[c] * rsqrtf(var + BN_EPS);
        float bi   = b3[c] - mean * s;
        out[i] = fmaxf(s * t3[i] + bi + x[i], 0.f);
    }
}

extern "C" void kernel_launch(void* const* d_in, const int* in_sizes, int n_in,
                              void* d_out, int out_size, void* d_ws, size_t ws_size,
                              hipStream_t stream)
{
    const float* p   = (const float*)d_in[0];
    const float* x   = (const float*)d_in[1];
    const int*   knn = (const int*)  d_in[2];
    const float* W1  = (const float*)d_in[3];
    const float* Wv  = (const float*)d_in[4];
    const float* Wp1 = (const float*)d_in[5];
    const float* Wp2 = (const float*)d_in[6];
    const float* Ww  = (const float*)d_in[7];
    const float* W3  = (const float*)d_in[8];
    const float* g1  = (const float*)d_in[9];
    const float* b1  = (const float*)d_in[10];
    const float* g2  = (const float*)d_in[11];
    const float* b2  = (const float*)d_in[12];
    const float* g3  = (const float*)d_in[13];
    const float* b3  = (const float*)d_in[14];
    float* out = (float*)d_out;

    const int N = in_sizes[0] / 3;          // p is [N,3]

    float* t1     = (float*)d_ws;           // N*64
    float* v      = t1 + (size_t)N * 64;    // N*64
    float* mixed  = v  + (size_t)N * 64;    // N*64
    float* stats  = mixed + (size_t)N * 64; // 3*128 floats
    float* stats1 = stats;
    float* stats2 = stats + 128;
    float* stats3 = stats + 256;
    float* t3     = t1;                     // t1 is dead after mixer -> reuse

    zero_stats_kernel<<<1, 384, 0, stream>>>(stats, 384);

    const int gemmGrid = (N + 63) / 64;
    // t1 = x @ W1, accumulate stats1
    gemm64_bn_kernel<<<gemmGrid, 128, 0, stream>>>(
        x, W1, t1, N, nullptr, nullptr, nullptr, stats1);
    // v = relu(BN1(t1)) @ Wv
    gemm64_bn_kernel<<<gemmGrid, 128, 0, stream>>>(
        t1, Wv, v, N, stats1, g1, b1, nullptr);
    // mixed = PointMixer intra-set layer, accumulate stats2
    mixer_kernel<<<(N + 3) / 4, 128, 0, stream>>>(
        p, knn, t1, v, Wp1, Wp2, Ww, stats1, g1, b1, mixed, stats2, N);
    // t3 = relu(BN2(mixed)) @ W3, accumulate stats3
    gemm64_bn_kernel<<<gemmGrid, 128, 0, stream>>>(
        mixed, W3, t3, N, stats2, g2, b2, stats3);
    // out = relu(BN3(t3) + x)
    final_kernel<<<(N * 64 + 255) / 256, 256, 0, stream>>>(
        t3, x, stats3, g3, b3, out, N);
}